// SoftMoE_29016799052043
// MI455X (gfx1250) — compile-verified
//
#include <hip/hip_runtime.h>
#include <math.h>
#include <stdint.h>

// MoE: D=1024, E=8, F=4096, tokens = B*S = 4096, top_k = 2.
typedef __attribute__((ext_vector_type(2))) float v2f;
typedef __attribute__((ext_vector_type(8))) float v8f;
typedef __attribute__((ext_vector_type(4))) unsigned int u32x4;
typedef __attribute__((ext_vector_type(4))) int i32x4;
typedef __attribute__((ext_vector_type(8))) int i32x8;

#define D_DIM 1024
#define E_NUM 8
#define F_DIM 4096
#define NTOK  4096
#define MAXROWS 4096   // max (token,expert) pairs per expert
#define KC 32          // K-chunk staged to LDS by the TDM
#define BN 128         // block N tile

// ---------------- Router: logits -> top2 -> softmax -> compacted expert lists
__global__ void router_kernel(const float* __restrict__ x,
                              const float* __restrict__ Wr,
                              const float* __restrict__ br,
                              int* __restrict__ counts,
                              int* __restrict__ lists,
                              float* __restrict__ gates) {
    int t = blockIdx.x * blockDim.x + threadIdx.x;
    if (t >= NTOK) return;
    float acc[E_NUM];
    #pragma unroll
    for (int e = 0; e < E_NUM; ++e) acc[e] = br[e];
    const float* xr = x + (size_t)t * D_DIM;
    for (int d = 0; d < D_DIM; ++d) {
        float xv = xr[d];
        const float* wrow = Wr + (size_t)d * E_NUM;
        #pragma unroll
        for (int e = 0; e < E_NUM; ++e) acc[e] = fmaf(xv, wrow[e], acc[e]);
    }
    // top-2 (first index wins ties, like lax.top_k)
    int i0 = 0; float v0 = acc[0];
    #pragma unroll
    for (int e = 1; e < E_NUM; ++e) if (acc[e] > v0) { v0 = acc[e]; i0 = e; }
    int i1 = -1; float v1 = -3.4e38f;
    #pragma unroll
    for (int e = 0; e < E_NUM; ++e) if (e != i0 && acc[e] > v1) { v1 = acc[e]; i1 = e; }
    float e1 = expf(v1 - v0);
    float inv = 1.0f / (1.0f + e1);

    int s0 = atomicAdd(&counts[i0], 1);
    lists[i0 * MAXROWS + s0] = t * 2 + 0;
    gates[i0 * MAXROWS + s0] = inv;
    int s1 = atomicAdd(&counts[i1], 1);
    lists[i1 * MAXROWS + s1] = t * 2 + 1;
    gates[i1 * MAXROWS + s1] = e1 * inv;
}

// ---------------- Tensor Data Mover: stage a KC x BN f32 weight tile into LDS.
// D# built per CDNA5 ISA ch.8: group0 = {count|flags, lds_addr, global_addr, type=2},
// group1 = {data_size=4B, tensor_dim0/1, tile_dim0/1, tensor_dim0_stride}.
__device__ __forceinline__ void tdm_load_tile(unsigned lds_byte_addr,
                                              const float* gsrc,
                                              unsigned rows, unsigned cols,
                                              unsigned row_stride_elems,
                                              unsigned tensor_rows) {
    unsigned long long ga = (unsigned long long)(uintptr_t)gsrc;
    u32x4 g0;
    g0[0] = 1u;                                   // count=1, user descriptor
    g0[1] = lds_byte_addr;                        // bits 63:32  lds_addr
    g0[2] = (unsigned)(ga & 0xFFFFFFFFu);         // bits 95:64  global_addr lo
    g0[3] = (unsigned)((ga >> 32) & 0x01FFFFFFu)  // bits 120:96 global_addr hi
          | (2u << 30);                           // bits 127:126 type=2 (image)
    i32x8 g1;
    g1[0] = (int)(2u << 16);                      // data_size=2 -> 4 bytes
    g1[1] = (int)((row_stride_elems & 0xFFFFu) << 16);            // tensor_dim0 lo
    g1[2] = (int)((row_stride_elems >> 16) |
                  ((tensor_rows & 0xFFFFu) << 16));               // dim0 hi | dim1 lo
    g1[3] = (int)((tensor_rows >> 16) | (cols << 16));            // dim1 hi | tile_dim0
    g1[4] = (int)rows;                                            // tile_dim1 (tile_dim2=0)
    g1[5] = (int)row_stride_elems;                                // tensor_dim0_stride lo
    g1[6] = 0;                                                    // stride hi | dim1_stride lo
    g1[7] = 0;
    i32x4 z4 = {0, 0, 0, 0};
#if defined(__clang_major__) && (__clang_major__ >= 23)
    i32x8 z8 = {0, 0, 0, 0, 0, 0, 0, 0};
    __builtin_amdgcn_tensor_load_to_lds(g0, g1, z4, z4, z8, 0);
#else
    __builtin_amdgcn_tensor_load_to_lds(g0, g1, z4, z4, 0);
#endif
}

// ---------------- Fused tiled GEMM body shared by fc1/fc2.
// Block: 256 threads = 8 waves (2x4), tile M=32 x N=128, each wave owns two
// 16x16 f32 WMMA accumulators. B tiles are TDM-staged into double-buffered LDS.
struct GemmFrag {
    v8f acc0, acc1;
};

template <int KTOT>
__device__ __forceinline__ GemmFrag gemm_body(const float* __restrict__ aRow,
                                              const float* __restrict__ gB,
                                              unsigned row_stride_elems,
                                              float* ldsB,   // 2 * KC * BN floats
                                              int wave, int khalf,
                                              int n0, int n1) {
    GemmFrag r;
    r.acc0 = (v8f){};
    r.acc1 = (v8f){};
    const int nk = KTOT / KC;
    const unsigned lds_base = (unsigned)(uintptr_t)ldsB;

    if (wave == 0)
        tdm_load_tile(lds_base, gB, KC, BN, row_stride_elems, KTOT);

    for (int kc = 0; kc < nk; ++kc) {
        if (wave == 0) {
            if (kc + 1 < nk) {
                tdm_load_tile(lds_base + ((kc + 1) & 1) * (KC * BN * 4u),
                              gB + (size_t)(kc + 1) * KC * row_stride_elems,
                              KC, BN, row_stride_elems, KTOT);
                __builtin_amdgcn_s_wait_tensorcnt(1);  // oldest tile landed
            } else {
                __builtin_amdgcn_s_wait_tensorcnt(0);
            }
        }
        __syncthreads();                               // release tile to all waves

        const float* bt = ldsB + (kc & 1) * (KC * BN);
        const float* ak = aRow + kc * KC;
        __builtin_prefetch(ak + 2 * KC, 0, 1);         // A stream hint
        #pragma unroll
        for (int kk = 0; kk < KC; kk += 4) {
            v2f a = *(const v2f*)(ak + kk);
            const float* bp = bt + (kk + khalf * 2) * BN;
            v2f b0, b1;
            b0.x = bp[n0];  b0.y = bp[BN + n0];
            b1.x = bp[n1];  b1.y = bp[BN + n1];
            r.acc0 = __builtin_amdgcn_wmma_f32_16x16x4_f32(
                false, a, false, b0, (short)0, r.acc0, false, false);
            r.acc1 = __builtin_amdgcn_wmma_f32_16x16x4_f32(
                false, a, false, b1, (short)0, r.acc1, false, false);
        }
        __syncthreads();                               // done reading before overwrite
    }
    return r;
}

// ---------------- FC1 + exact GELU:  h[pair][F] = gelu(x[token] @ W1[e] + b1[e])
__global__ void __launch_bounds__(256)
moe_fc1_kernel(const float* __restrict__ x,
               const float* __restrict__ W1,
               const float* __restrict__ b1,
               const int*   __restrict__ counts,
               const int*   __restrict__ lists,
               float* __restrict__ h) {
    __shared__ float ldsB[2 * KC * BN];

    const int e     = blockIdx.z;
    const int count = counts[e];
    const int m0    = blockIdx.x * 32;
    if (m0 >= count) return;

    const int lane  = threadIdx.x & 31;
    const int wave  = threadIdx.x >> 5;
    const int wrow  = wave >> 2;            // 0..1
    const int wcol  = wave & 3;             // 0..3
    const int khalf = lane >> 4;            // ISA 16x4 A-fragment K-half
    const int l16   = lane & 15;
    const int n0    = wcol * 32 + l16;
    const int n1    = n0 + 16;

    const int mg   = m0 + wrow * 16 + l16;
    const int pair = (mg < count) ? lists[e * MAXROWS + mg] : 0;
    const float* aRow = x + (size_t)(pair >> 1) * D_DIM + khalf * 2;

    const float* gB = W1 + (size_t)e * D_DIM * F_DIM + (size_t)blockIdx.y * BN;

    GemmFrag r = gemm_body<D_DIM>(aRow, gB, F_DIM, ldsB, wave, khalf, n0, n1);

    const int ng0 = blockIdx.y * BN + n0;
    const int ng1 = blockIdx.y * BN + n1;
    const float bias0 = b1[e * F_DIM + ng0];
    const float bias1 = b1[e * F_DIM + ng1];
    #pragma unroll
    for (int j = 0; j < 8; ++j) {
        int mgs = m0 + wrow * 16 + khalf * 8 + j;     // C/D VGPR row layout
        if (mgs < count) {
            int pr = lists[e * MAXROWS + mgs];
            float va = r.acc0[j] + bias0;
            float vb = r.acc1[j] + bias1;
            h[(size_t)pr * F_DIM + ng0] = 0.5f * va * (1.0f + erff(va * 0.70710678118f));
            h[(size_t)pr * F_DIM + ng1] = 0.5f * vb * (1.0f + erff(vb * 0.70710678118f));
        }
    }
}

// ---------------- FC2 + gated scatter:  out[token] += gate * (h[pair] @ W2[e] + b2[e])
__global__ void __launch_bounds__(256)
moe_fc2_kernel(const float* __restrict__ h,
               const float* __restrict__ W2,
               const float* __restrict__ b2,
               const int*   __restrict__ counts,
               const int*   __restrict__ lists,
               const float* __restrict__ gates,
               float* __restrict__ out) {
    __shared__ float ldsB[2 * KC * BN];

    const int e     = blockIdx.z;
    const int count = counts[e];
    const int m0    = blockIdx.x * 32;
    if (m0 >= count) return;

    const int lane  = threadIdx.x & 31;
    const int wave  = threadIdx.x >> 5;
    const int wrow  = wave >> 2;
    const int wcol  = wave & 3;
    const int khalf = lane >> 4;
    const int l16   = lane & 15;
    const int n0    = wcol * 32 + l16;
    const int n1    = n0 + 16;

    const int mg   = m0 + wrow * 16 + l16;
    const int pair = (mg < count) ? lists[e * MAXROWS + mg] : 0;
    const float* aRow = h + (size_t)pair * F_DIM + khalf * 2;

    const float* gB = W2 + (size_t)e * F_DIM * D_DIM + (size_t)blockIdx.y * BN;

    GemmFrag r = gemm_body<F_DIM>(aRow, gB, D_DIM, ldsB, wave, khalf, n0, n1);

    const int ng0 = blockIdx.y * BN + n0;
    const int ng1 = blockIdx.y * BN + n1;
    const float bias0 = b2[e * D_DIM + ng0];
    const float bias1 = b2[e * D_DIM + ng1];
    #pragma unroll
    for (int j = 0; j < 8; ++j) {
        int mgs = m0 + wrow * 16 + khalf * 8 + j;
        if (mgs < count) {
            int pr   = lists[e * MAXROWS + mgs];
            float gw = gates[e * MAXROWS + mgs];
            size_t ob = (size_t)(pr >> 1) * D_DIM;
            atomicAdd(&out[ob + ng0], (r.acc0[j] + bias0) * gw);
            atomicAdd(&out[ob + ng1], (r.acc1[j] + bias1) * gw);
        }
    }
}

extern "C" void kernel_launch(void* const* d_in, const int* in_sizes, int n_in,
                              void* d_out, int out_size, void* d_ws, size_t ws_size,
                              hipStream_t stream) {
    const float* x  = (const float*)d_in[0];
    const float* Wr = (const float*)d_in[1];
    const float* br = (const float*)d_in[2];
    const float* W1 = (const float*)d_in[3];
    const float* b1 = (const float*)d_in[4];
    const float* W2 = (const float*)d_in[5];
    const float* b2 = (const float*)d_in[6];
    // d_in[7] = top_k (== 2, baked into the kernels)
    float* out = (float*)d_out;

    char*  ws     = (char*)d_ws;
    int*   counts = (int*)ws;                                   // 8 ints
    int*   lists  = (int*)(ws + 1024);                          // 8 x 4096 ints
    float* gates  = (float*)(ws + 1024 + E_NUM * MAXROWS * 4);  // 8 x 4096 floats
    float* h      = (float*)(ws + (1u << 20));                  // 8192 x 4096 floats

    hipMemsetAsync(counts, 0, E_NUM * sizeof(int), stream);
    hipMemsetAsync(d_out, 0, (size_t)out_size * sizeof(float), stream);

    router_kernel<<<NTOK / 256, 256, 0, stream>>>(x, Wr, br, counts, lists, gates);

    dim3 g1(NTOK / 32, F_DIM / BN, E_NUM);   // 128 x 32 x 8
    moe_fc1_kernel<<<g1, 256, 0, stream>>>(x, W1, b1, counts, lists, h);

    dim3 g2(NTOK / 32, D_DIM / BN, E_NUM);   // 128 x 8 x 8
    moe_fc2_kernel<<<g2, 256, 0, stream>>>(h, W2, b2, counts, lists, gates, out);
}